// TaskProcessor_45921790329019
// MI455X (gfx1250) — compile-verified
//
#include <hip/hip_runtime.h>
#include <hip/hip_bf16.h>
#include <math.h>

// Problem constants (from reference)
#define BB 32
#define PP 2048
#define DD 768
#define TT 3
#define OO 128
#define HH 192
#define BPp (BB * PP)   // 65536 tokens

typedef __attribute__((ext_vector_type(2))) float v2f;
typedef __attribute__((ext_vector_type(8))) float v8f;

// fp32 WMMA: D(16x16 f32) = A(16x4 f32) * B(4x16 f32) + C
__device__ __forceinline__ v8f wmma4(v2f a, v2f b, v8f c) {
    return __builtin_amdgcn_wmma_f32_16x16x4_f32(
        /*neg_a=*/false, a, /*neg_b=*/false, b,
        /*c_mod=*/(short)0, c, /*reuse_a=*/false, /*reuse_b=*/false);
}

__device__ __forceinline__ float gelu_exact(float x) {
    return 0.5f * x * (1.0f + erff(x * 0.70710678118654752f));
}

// ---------------------------------------------------------------------------
// K1: tiny precompute. q[t] = tt[t]@Wq[t]+bq ; kq_scaled[t] = scale*(Wk[t]@q[t]);
//     sbias[t] = scale*(q[t]·bk[t]).  One block per task.
// ---------------------------------------------------------------------------
__global__ void precompute_kernel(const float* __restrict__ tt,
                                  const float* __restrict__ Wq, const float* __restrict__ bq,
                                  const float* __restrict__ Wk, const float* __restrict__ bk,
                                  float* __restrict__ kq_scaled, float* __restrict__ sbias) {
    const int t = blockIdx.x;
    const int tid = threadIdx.x;
    __shared__ float qv[DD];
    __shared__ float red[256];
    const float scale = 0.03608439182435161f; // 1/sqrt(768)

    for (int e = tid; e < DD; e += 256) {
        float s = bq[t * DD + e];
        const float* w = Wq + (size_t)t * DD * DD + e;
        const float* tr = tt + t * DD;
        for (int d = 0; d < DD; ++d) s += tr[d] * w[(size_t)d * DD];
        qv[e] = s;
    }
    __syncthreads();

    float sb = 0.f;
    for (int e = tid; e < DD; e += 256) sb += qv[e] * bk[t * DD + e];
    red[tid] = sb; __syncthreads();
    for (int s = 128; s > 0; s >>= 1) { if (tid < s) red[tid] += red[tid + s]; __syncthreads(); }
    if (tid == 0) sbias[t] = scale * red[0];

    for (int d = tid; d < DD; d += 256) {
        const float* w = Wk + (size_t)t * DD * DD + (size_t)d * DD;
        float s = 0.f;
        for (int e = 0; e < DD; ++e) s += w[e] * qv[e];
        kq_scaled[t * DD + d] = scale * s;
    }
}

// ---------------------------------------------------------------------------
// K2: scores[t, tok] = x[tok]·kq_scaled[t] + sbias[t]  via fp32 WMMA.
//     Per wave: 16 tokens (A-tile), B = kq vectors in cols 0..2 (padded to 16),
//     staged in LDS. Grid: BPp/64 blocks of 4 waves.
// ---------------------------------------------------------------------------
__global__ void scores_kernel(const float* __restrict__ x,
                              const float* __restrict__ kq_scaled,
                              const float* __restrict__ sbias,
                              float* __restrict__ scores) {
    __shared__ float KQ[DD * 16]; // 48 KB: B-matrix, row k, col n
    const int tid = threadIdx.x;
    for (int i = tid; i < DD * 16; i += 128) {
        int k = i >> 4, n = i & 15;
        KQ[i] = (n < TT) ? kq_scaled[n * DD + k] : 0.f;
    }
    __syncthreads();

    const int wave = tid >> 5, l = tid & 31;
    const int half = l >> 4, lm = l & 15;
    const long tok0 = (long)blockIdx.x * 64 + wave * 16;
    const float* xrow = x + (tok0 + lm) * (long)DD;

    v8f c = {};
    for (int kk = 0; kk < DD; kk += 4) {
        v2f a = *(const v2f*)(xrow + kk + half * 2);            // A: row=token, K pair
        v2f b;
        b[0] = KQ[(kk + half * 2) * 16 + lm];                   // B: K row, col=lane
        b[1] = KQ[(kk + half * 2 + 1) * 16 + lm];
        c = wmma4(a, b, c);
    }
    // C element (m,n): VGPR r -> m = r + 8*half, n = lm. Keep cols 0..2.
    if (lm < TT) {
        const float sb = sbias[lm];
        float* sp = scores + (size_t)lm * BPp + tok0 + half * 8;
        #pragma unroll
        for (int r = 0; r < 8; ++r) sp[r] = c[r] + sb;
    }
}

// ---------------------------------------------------------------------------
// K3: masked softmax over P per (t,b) row, in place on scores buffer.
// ---------------------------------------------------------------------------
__global__ void softmax_kernel(float* __restrict__ scores,
                               const unsigned char* __restrict__ mask) {
    const int b = blockIdx.x, t = blockIdx.y;
    const int tid = threadIdx.x;
    float* sc = scores + (size_t)t * BPp + (size_t)b * PP;
    const unsigned char* mk = mask + (size_t)b * PP;
    float v[8];
    float mx = -3.402823e38f;
    #pragma unroll
    for (int i = 0; i < 8; ++i) {
        int p = tid + i * 256;
        float s = sc[p];
        if (mk[p]) s = -1e9f;
        v[i] = s;
        mx = fmaxf(mx, s);
    }
    __shared__ float red[256];
    red[tid] = mx; __syncthreads();
    for (int s = 128; s > 0; s >>= 1) { if (tid < s) red[tid] = fmaxf(red[tid], red[tid + s]); __syncthreads(); }
    mx = red[0]; __syncthreads();
    float sum = 0.f;
    #pragma unroll
    for (int i = 0; i < 8; ++i) { v[i] = expf(v[i] - mx); sum += v[i]; }
    red[tid] = sum; __syncthreads();
    for (int s = 128; s > 0; s >>= 1) { if (tid < s) red[tid] += red[tid + s]; __syncthreads(); }
    const float inv = 1.f / red[0];
    #pragma unroll
    for (int i = 0; i < 8; ++i) sc[tid + i * 256] = v[i] * inv;
}

// ---------------------------------------------------------------------------
// K4: xbar[t,b,:] = sum_p attn[t,b,p] * x[b,p,:]  via fp32 WMMA.
//     A = attn (3 task rows padded to 16) x 4 p's; B = x 4 p's x 16 dims.
//     Grid: (D/64, B), block = 4 waves; wave handles one 16-dim chunk.
// ---------------------------------------------------------------------------
__global__ void pool_kernel(const float* __restrict__ x,
                            const float* __restrict__ attn,
                            float* __restrict__ xbar) {
    const int b = blockIdx.y;
    const int tid = threadIdx.x;
    const int wave = tid >> 5, l = tid & 31;
    const int half = l >> 4, lm = l & 15;
    const int d0 = (blockIdx.x * 4 + wave) * 16;
    const float* attb = attn + (size_t)b * PP;
    const float* xb = x + (size_t)b * PP * DD + d0 + lm;

    v8f c = {};
    for (int kk = 0; kk < PP; kk += 4) {
        const int kp = kk + half * 2;
        v2f a;
        if (lm < TT) a = *(const v2f*)(attb + (size_t)lm * BPp + kp);
        else { a[0] = 0.f; a[1] = 0.f; }
        v2f bb;
        bb[0] = xb[(size_t)kp * DD];
        bb[1] = xb[(size_t)(kp + 1) * DD];
        c = wmma4(a, bb, c);
    }
    if (half == 0) {  // rows 0..2 live in VGPRs 0..2 of the low half
        #pragma unroll
        for (int r = 0; r < TT; ++r)
            xbar[(size_t)r * BB * DD + (size_t)b * DD + d0 + lm] = c[r];
    }
}

// ---------------------------------------------------------------------------
// K5: generic batched (over t) WMMA fp32 GEMM with fused bias/residual/GELU.
//     One wave per 16x16 output tile. C = A(MxK) @ B(KxN) + bias[n] (+resid[n]).
// ---------------------------------------------------------------------------
__global__ void gemm_wmma(const float* __restrict__ A, const float* __restrict__ Bm,
                          const float* __restrict__ bias, const float* __restrict__ resid,
                          float* __restrict__ C,
                          int M, int N, int K,
                          long bsA, long bsB, long bsBias, long bsResid, long bsC,
                          int act) {
    const int t = blockIdx.z;
    const int n0 = blockIdx.x * 16, m0 = blockIdx.y * 16;
    const int l = threadIdx.x;
    const int half = l >> 4, lm = l & 15;
    const float* At = A + (size_t)t * bsA + (size_t)(m0 + lm) * K;
    const float* Bt = Bm + (size_t)t * bsB + n0 + lm;

    v8f c = {};
    for (int kk = 0; kk < K; kk += 4) {
        v2f a = *(const v2f*)(At + kk + half * 2);
        v2f b;
        b[0] = Bt[(size_t)(kk + half * 2) * N];
        b[1] = Bt[(size_t)(kk + half * 2 + 1) * N];
        c = wmma4(a, b, c);
    }
    float bc = bias[(size_t)t * bsBias + n0 + lm];
    if (resid) bc += resid[(size_t)t * bsResid + n0 + lm];
    float* Ct = C + (size_t)t * bsC + (size_t)(m0 + half * 8) * N + n0 + lm;
    #pragma unroll
    for (int r = 0; r < 8; ++r) {
        float v = c[r] + bc;
        if (act == 1) v = gelu_exact(v);
        Ct[(size_t)r * N] = v;
    }
}

// ---------------------------------------------------------------------------
extern "C" void kernel_launch(void* const* d_in, const int* in_sizes, int n_in,
                              void* d_out, int out_size, void* d_ws, size_t ws_size,
                              hipStream_t stream) {
    const float* x   = (const float*)d_in[0];
    const unsigned char* mask = (const unsigned char*)d_in[1]; // numpy bool = 1 byte
    const float* tt  = (const float*)d_in[2];
    const float* Wq  = (const float*)d_in[3];
    const float* bq  = (const float*)d_in[4];
    const float* Wk  = (const float*)d_in[5];
    const float* bk  = (const float*)d_in[6];
    const float* Wv  = (const float*)d_in[7];
    const float* bv  = (const float*)d_in[8];
    const float* Wo  = (const float*)d_in[9];
    const float* bo  = (const float*)d_in[10];
    const float* h1w = (const float*)d_in[11];
    const float* h1b = (const float*)d_in[12];
    const float* h2w = (const float*)d_in[13];
    const float* h2b = (const float*)d_in[14];
    float* out = (float*)d_out;
    float* ws  = (float*)d_ws;

    // Workspace layout (floats)
    float* kq     = ws + 0;        // T*D          = 2304
    float* sbias  = ws + 2304;     // T (padded)
    float* scores = ws + 2560;     // T*B*P        = 196608 (reused as attn)
    float* xbar   = ws + 199168;   // T*B*D        = 73728
    float* y1     = ws + 272896;   // T*B*D
    float* y2     = ws + 346624;   // T*B*D
    float* hbuf   = ws + 420352;   // T*B*H        = 18432

    precompute_kernel<<<TT, 256, 0, stream>>>(tt, Wq, bq, Wk, bk, kq, sbias);

    scores_kernel<<<BPp / 64, 128, 0, stream>>>(x, kq, sbias, scores);

    softmax_kernel<<<dim3(BB, TT), 256, 0, stream>>>(scores, mask);

    pool_kernel<<<dim3(DD / 64, BB), 128, 0, stream>>>(x, scores, xbar);

    // y1 = xbar @ Wv + bv
    gemm_wmma<<<dim3(DD / 16, BB / 16, TT), 32, 0, stream>>>(
        xbar, Wv, bv, nullptr, y1, BB, DD, DD,
        (long)BB * DD, (long)DD * DD, DD, 0, (long)BB * DD, 0);
    // y2 = y1 @ Wo + bo + task_token (residual broadcast over rows)
    gemm_wmma<<<dim3(DD / 16, BB / 16, TT), 32, 0, stream>>>(
        y1, Wo, bo, tt, y2, BB, DD, DD,
        (long)BB * DD, (long)DD * DD, DD, DD, (long)BB * DD, 0);
    // h = gelu(y2 @ h1w + h1b)
    gemm_wmma<<<dim3(HH / 16, BB / 16, TT), 32, 0, stream>>>(
        y2, h1w, h1b, nullptr, hbuf, BB, HH, DD,
        (long)BB * DD, (long)DD * HH, HH, 0, (long)BB * HH, 1);
    // out = h @ h2w + h2b  -> (T,B,O) directly into d_out
    gemm_wmma<<<dim3(OO / 16, BB / 16, TT), 32, 0, stream>>>(
        hbuf, h2w, h2b, nullptr, out, BB, OO, HH,
        (long)BB * HH, (long)HH * OO, OO, 0, (long)BB * OO, 0);
}